// Generator_69578470195671
// MI455X (gfx1250) — compile-verified
//
#include <hip/hip_runtime.h>
#include <math.h>
#include <stdint.h>

// ---------------- CDNA5 WMMA (full-precision f32, 16x16x4) ----------------
typedef __attribute__((ext_vector_type(2))) float v2f;
typedef __attribute__((ext_vector_type(8))) float v8f;

__device__ __forceinline__ v8f wmma4(v2f a, v2f b, v8f c) {
  return __builtin_amdgcn_wmma_f32_16x16x4_f32(false, a, false, b, (short)0, c,
                                               false, false);
}

// Async global -> LDS copy (16B per lane), tracked by ASYNCcnt.
__device__ __forceinline__ void async_b128(unsigned lds_byte_off, const float* gptr) {
  asm volatile("global_load_async_to_lds_b128 %0, %1, off"
               :: "v"(lds_byte_off), "v"(gptr)
               : "memory");
}
__device__ __forceinline__ void wait_async0() {
  asm volatile("s_wait_asynccnt 0x0" ::: "memory");
}

// Problem constants (from setup_inputs)
#define BB 64
#define TT 12
#define NN 1024
#define DD 16
#define HH 64
#define CI 65            // C_IN + H
#define NC (BB * CI)     // 4160 columns in the mixing GEMM
#define KI 195           // 3 * 65
#define KIP 196          // padded to multiple of 4
#define OG 128
#define OU 64

// ---------------- 1) layernorms + per-node biases ----------------
__global__ void prep_kernel(const float* __restrict__ node_emb,
                            const float* __restrict__ time_emb_t,
                            const float* __restrict__ gln_g, const float* __restrict__ gln_b,
                            const float* __restrict__ uln_g, const float* __restrict__ uln_b,
                            const float* __restrict__ gate_b, const float* __restrict__ upd_b,
                            float* __restrict__ ne_g, float* __restrict__ ne_u,
                            float* __restrict__ bias_g, float* __restrict__ bias_u) {
  __shared__ float emb_s[DD], neg_s[DD], neu_s[DD];
  int n = blockIdx.x;
  int tid = threadIdx.x;
  if (tid < DD) emb_s[tid] = node_emb[n * DD + tid] + time_emb_t[tid];
  __syncthreads();
  float mu = 0.f;
  for (int d = 0; d < DD; ++d) mu += emb_s[d];
  mu *= (1.f / DD);
  float var = 0.f;
  for (int d = 0; d < DD; ++d) { float df = emb_s[d] - mu; var += df * df; }
  var *= (1.f / DD);
  float inv = rsqrtf(var + 1e-12f);
  if (tid < DD) {
    float xc = (emb_s[tid] - mu) * inv;
    float vg = xc * gln_g[tid] + gln_b[tid];
    float vu = xc * uln_g[tid] + uln_b[tid];
    neg_s[tid] = vg; neu_s[tid] = vu;
    ne_g[n * DD + tid] = vg; ne_u[n * DD + tid] = vu;
  }
  __syncthreads();
  {                       // bias_g[n, o], o = 0..127   (blockDim.x == 128)
    float s = 0.f;
    for (int d = 0; d < DD; ++d) s += neg_s[d] * gate_b[d * OG + tid];
    bias_g[n * OG + tid] = s;
  }
  if (tid < OU) {         // bias_u[n, o], o = 0..63
    float s = 0.f;
    for (int d = 0; d < DD; ++d) s += neu_s[d] * upd_b[d * OU + tid];
    bias_u[n * OU + tid] = s;
  }
}

// ---------------- 2) adaptive adjacency: row softmax of ne @ ne^T ----------------
__global__ void __launch_bounds__(256)
adjacency_kernel(const float* __restrict__ ne_g, const float* __restrict__ ne_u,
                 float* __restrict__ S_g, float* __restrict__ S_u) {
  int bid = blockIdx.x;
  int which = bid >> 10;
  int n = bid & (NN - 1);
  const float* ne = which ? ne_u : ne_g;
  float* S = which ? S_u : S_g;
  __shared__ float red[256];
  __shared__ float row[DD];
  int tid = threadIdx.x;
  if (tid < DD) row[tid] = ne[n * DD + tid];
  __syncthreads();
  float lg[4];
  float lmax = -3.0e38f;
  for (int j = 0; j < 4; ++j) {
    int m = tid + 256 * j;
    float s = 0.f;
    for (int d = 0; d < DD; ++d) s += row[d] * ne[m * DD + d];
    lg[j] = s; lmax = fmaxf(lmax, s);
  }
  red[tid] = lmax; __syncthreads();
  for (int off = 128; off > 0; off >>= 1) {
    if (tid < off) red[tid] = fmaxf(red[tid], red[tid + off]);
    __syncthreads();
  }
  float gmax = red[0]; __syncthreads();
  float lsum = 0.f;
  for (int j = 0; j < 4; ++j) { lg[j] = __expf(lg[j] - gmax); lsum += lg[j]; }
  red[tid] = lsum; __syncthreads();
  for (int off = 128; off > 0; off >>= 1) {
    if (tid < off) red[tid] += red[tid + off];
    __syncthreads();
  }
  float rinv = 1.f / red[0];
  for (int j = 0; j < 4; ++j) S[(size_t)n * NN + tid + 256 * j] = lg[j] * rinv;
}

// ---------------- 3) build concatenated features Z[m, b*65+c] ----------------
__global__ void build_cat_kernel(const float* __restrict__ x, const float* __restrict__ out_hist,
                                 const float* __restrict__ zr, float* __restrict__ Z,
                                 int t, int mode, int prev_valid) {
  int idx = blockIdx.x * blockDim.x + threadIdx.x;
  if (idx >= NN * NC) return;
  int m = idx / NC;
  int col = idx - m * NC;
  int b = col / CI;
  int c = col - b * CI;
  float v;
  if (c == 0) {
    v = x[((size_t)b * TT + t) * NN + m];
  } else {
    int j = c - 1;
    float st = prev_valid ? out_hist[(((size_t)b * TT + (t - 1)) * NN + m) * HH + j] : 0.f;
    if (mode) st *= zr[((size_t)m * BB + b) * OG + j];   // z gate
    v = st;
  }
  Z[idx] = v;
}

// ---------------- 4) LDS-blocked graph-mix GEMM with async double buffering -------
// Y = S@Bm (mode 0) | 2*S@Bm - X (mode 1).  Macro-tile 128x128, Kc = 32.
#define APITCH 36                 // dwords; banks: gcd(36,64)=4 -> 16 rows conflict-free
#define BPITCH 132                // dwords
#define A_DW (128 * APITCH)       // 4608 dwords per A buffer
#define B_DW (32 * BPITCH)        // 4224 dwords per B buffer

__global__ void __launch_bounds__(256)
gemm_mix_kernel(const float* __restrict__ S, const float* __restrict__ Bm,
                const float* __restrict__ X, float* __restrict__ Y, int mode) {
  extern __shared__ float lds[];

  const int wid = threadIdx.x >> 5;
  const int lane = threadIdx.x & 31;
  const int half = lane >> 4;
  const int l16 = lane & 15;

  const int bm = blockIdx.x / 33;      // 8 m-blocks
  const int bn = blockIdx.x - bm * 33; // 33 n-blocks (last partially valid)
  const int m0 = bm * 128;
  const int n0 = bn * 128;

  // Per-wave async slice bases (A: 4 row-quads; B: 4 rows per slab)
  const int arow = wid * 16 + (lane >> 3);          // + 4*j
  const int achunk = (lane & 7) * 4;                // dword within row
  const int brow = wid * 4;                         // + j
  const unsigned aLdsRow = (unsigned)((arow * APITCH + achunk) * 4);
  const unsigned bLdsRow = (unsigned)((brow * BPITCH + lane * 4) * 4);
  // Byte offset of the dynamic-LDS block (generic LDS address low 32 bits
  // are the LDS offset, ISA sec.10.2) -- runtime cast, NOT a static initializer.
  const unsigned ldsBase = (unsigned)(uintptr_t)(void*)lds;

  const int wm2 = wid & 3;            // m-tile pair 0..3
  const int wn4 = wid >> 2;           // n-tile quad 0..1

  v8f acc[2][4];
#pragma unroll
  for (int a = 0; a < 2; ++a)
#pragma unroll
    for (int c = 0; c < 4; ++c)
      acc[a][c] = (v8f){0.f, 0.f, 0.f, 0.f, 0.f, 0.f, 0.f, 0.f};

  // ---- async issue of one 32-wide K slab into buffer `buf` ----
  auto issue = [&](int k0, int buf) {
    unsigned aB = ldsBase + (unsigned)(buf * A_DW * 4);
    unsigned bB = ldsBase + (unsigned)((2 * A_DW + buf * B_DW) * 4);
#pragma unroll
    for (int j = 0; j < 4; ++j) {   // A: rows (wid*16 + 4j .. +3), 32 cols
      const float* g = S + (size_t)(m0 + arow + 4 * j) * NN + k0 + achunk;
      async_b128(aB + aLdsRow + (unsigned)(4 * j * APITCH * 4), g);
    }
#pragma unroll
    for (int j = 0; j < 4; ++j) {   // B: row k0+wid*4+j, 128 cols
      const float* g = Bm + (size_t)(k0 + brow + j) * NC + n0 + lane * 4;
      async_b128(bB + bLdsRow + (unsigned)(j * BPITCH * 4), g);
    }
  };

  issue(0, 0);
  wait_async0();
  __syncthreads();

  for (int ko = 0; ko < 32; ++ko) {
    int cur = ko & 1;
    if (ko + 1 < 32) issue((ko + 1) * 32, 1 - cur);

    const float* Ab = lds + cur * A_DW;                 // scalar LDS pointers
    const float* Bb = lds + 2 * A_DW + cur * B_DW;
#pragma unroll
    for (int ks = 0; ks < 8; ++ks) {
      int kb = ks * 4 + 2 * half;
      v2f af[2], bf[4];
#pragma unroll
      for (int a = 0; a < 2; ++a)
        af[a] = *(const v2f*)(Ab + ((wm2 * 2 + a) * 16 + l16) * APITCH + kb);
#pragma unroll
      for (int c = 0; c < 4; ++c) {
        int col = (wn4 * 4 + c) * 16 + l16;
        bf[c].x = Bb[kb * BPITCH + col];
        bf[c].y = Bb[(kb + 1) * BPITCH + col];
      }
#pragma unroll
      for (int a = 0; a < 2; ++a)
#pragma unroll
        for (int c = 0; c < 4; ++c)
          acc[a][c] = wmma4(af[a], bf[c], acc[a][c]);
    }

    if (ko + 1 < 32) {
      wait_async0();
      __syncthreads();
    }
  }

  // ---- epilogue ----
#pragma unroll
  for (int a = 0; a < 2; ++a) {
    int rbase = m0 + (wm2 * 2 + a) * 16 + 8 * half;
#pragma unroll
    for (int c = 0; c < 4; ++c) {
      int col = n0 + (wn4 * 4 + c) * 16 + l16;
      if (col < NC) {                      // wave-uniform per tile (NC % 16 == 0)
        if (mode) {
          for (int r = 0; r < 8; ++r) {
            size_t off = (size_t)(rbase + r) * NC + col;
            Y[off] = 2.f * acc[a][c][r] - X[off];
          }
        } else {
          for (int r = 0; r < 8; ++r)
            Y[(size_t)(rbase + r) * NC + col] = acc[a][c][r];
        }
      }
    }
  }
}

// ---------------- 5a) per-node filter GEMM + sigmoid (gate, O=128) ----------------
__global__ void __launch_bounds__(256)
pernode_gate_kernel(const float* __restrict__ Z, const float* __restrict__ Y1,
                    const float* __restrict__ Y2, const float* __restrict__ ne_g,
                    const float* __restrict__ gate_W, const float* __restrict__ bias_g,
                    float* __restrict__ zr) {
  extern __shared__ float lds[];
  float* A_s = lds;                   // [64][196]
  float* W_s = lds + BB * KIP;        // [196][128]
  int n = blockIdx.x;
  int tid = threadIdx.x;
  float ner[DD];
  for (int d = 0; d < DD; ++d) ner[d] = ne_g[n * DD + d];
  for (int idx = tid; idx < KIP * OG; idx += 256) {     // Wn = sum_d ne[d]*W[d]
    int ki = idx >> 7, o = idx & (OG - 1);
    float s = 0.f;
    if (ki < KI) {
      int k = ki / CI, i = ki - k * CI;
      const float* wp = gate_W + ((size_t)k * CI + i) * OG + o;
      for (int d = 0; d < DD; ++d) s += ner[d] * wp[(size_t)d * 3 * CI * OG];
    }
    W_s[idx] = s;
  }
  for (int idx = tid; idx < BB * KIP; idx += 256) {     // A panel (k-padded)
    int b = idx / KIP, ki = idx - b * KIP;
    float v = 0.f;
    if (ki < KI) {
      int k = ki / CI, i = ki - k * CI;
      const float* Fk = (k == 0) ? Z : ((k == 1) ? Y1 : Y2);
      v = Fk[(size_t)n * NC + b * CI + i];
    }
    A_s[idx] = v;
  }
  __syncthreads();
  int wid = tid >> 5, lane = tid & 31;
  int half = lane >> 4, l16 = lane & 15;
  for (int tt = wid; tt < 32; tt += 8) {     // 4 b-tiles x 8 o-tiles
    int tm = tt >> 3, to = tt & 7;
    v8f acc = {0.f, 0.f, 0.f, 0.f, 0.f, 0.f, 0.f, 0.f};
    const float* ap = A_s + (tm * 16 + l16) * KIP + 2 * half;
    const float* bp = W_s + (2 * half) * OG + to * 16 + l16;
#pragma unroll 7
    for (int kk = 0; kk < KIP / 4; ++kk) {
      v2f a = *(const v2f*)ap;
      v2f b; b.x = bp[0]; b.y = bp[OG];
      acc = wmma4(a, b, acc);
      ap += 4; bp += 4 * OG;
    }
    int o = to * 16 + l16;
    float bi = bias_g[n * OG + o];
    for (int r = 0; r < 8; ++r) {
      int b = tm * 16 + r + 8 * half;
      float v = acc[r] + bi;
      v = 1.f / (1.f + __expf(-v));          // sigmoid
      zr[((size_t)n * BB + b) * OG + o] = v;
    }
  }
}

// ---------------- 5b) per-node filter GEMM + tanh + GRU blend (update, O=64) --------
__global__ void __launch_bounds__(256)
pernode_update_kernel(const float* __restrict__ Z, const float* __restrict__ Y1,
                      const float* __restrict__ Y2, const float* __restrict__ ne_u,
                      const float* __restrict__ upd_W, const float* __restrict__ bias_u,
                      const float* __restrict__ zr, float* __restrict__ out,
                      int t, int prev_valid) {
  extern __shared__ float lds[];
  float* A_s = lds;                   // [64][196]
  float* W_s = lds + BB * KIP;        // [196][64]
  int n = blockIdx.x;
  int tid = threadIdx.x;
  float ner[DD];
  for (int d = 0; d < DD; ++d) ner[d] = ne_u[n * DD + d];
  for (int idx = tid; idx < KIP * OU; idx += 256) {
    int ki = idx >> 6, o = idx & (OU - 1);
    float s = 0.f;
    if (ki < KI) {
      int k = ki / CI, i = ki - k * CI;
      const float* wp = upd_W + ((size_t)k * CI + i) * OU + o;
      for (int d = 0; d < DD; ++d) s += ner[d] * wp[(size_t)d * 3 * CI * OU];
    }
    W_s[idx] = s;
  }
  for (int idx = tid; idx < BB * KIP; idx += 256) {
    int b = idx / KIP, ki = idx - b * KIP;
    float v = 0.f;
    if (ki < KI) {
      int k = ki / CI, i = ki - k * CI;
      const float* Fk = (k == 0) ? Z : ((k == 1) ? Y1 : Y2);
      v = Fk[(size_t)n * NC + b * CI + i];
    }
    A_s[idx] = v;
  }
  __syncthreads();
  int wid = tid >> 5, lane = tid & 31;
  int half = lane >> 4, l16 = lane & 15;
  for (int tt = wid; tt < 16; tt += 8) {     // 4 b-tiles x 4 o-tiles
    int tm = tt >> 2, to = tt & 3;
    v8f acc = {0.f, 0.f, 0.f, 0.f, 0.f, 0.f, 0.f, 0.f};
    const float* ap = A_s + (tm * 16 + l16) * KIP + 2 * half;
    const float* bp = W_s + (2 * half) * OU + to * 16 + l16;
#pragma unroll 7
    for (int kk = 0; kk < KIP / 4; ++kk) {
      v2f a = *(const v2f*)ap;
      v2f b; b.x = bp[0]; b.y = bp[OU];
      acc = wmma4(a, b, acc);
      ap += 4; bp += 4 * OU;
    }
    int o = to * 16 + l16;
    float bi = bias_u[n * OU + o];
    for (int r = 0; r < 8; ++r) {
      int b = tm * 16 + r + 8 * half;
      float hc = tanhf(acc[r] + bi);
      float rv = zr[((size_t)n * BB + b) * OG + OU + o];   // r gate
      float st = prev_valid ? out[(((size_t)b * TT + (t - 1)) * NN + n) * HH + o] : 0.f;
      out[(((size_t)b * TT + t) * NN + n) * HH + o] = rv * st + (1.f - rv) * hc;
    }
  }
}

// ---------------- driver ----------------
extern "C" void kernel_launch(void* const* d_in, const int* in_sizes, int n_in,
                              void* d_out, int out_size, void* d_ws, size_t ws_size,
                              hipStream_t stream) {
  (void)in_sizes; (void)n_in; (void)out_size; (void)ws_size;
  const float* x        = (const float*)d_in[0];
  const float* node_emb = (const float*)d_in[1];
  const float* time_emb = (const float*)d_in[2];
  const float* gate_W   = (const float*)d_in[3];
  const float* gate_b   = (const float*)d_in[4];
  const float* gln_g    = (const float*)d_in[5];
  const float* gln_b    = (const float*)d_in[6];
  const float* upd_W    = (const float*)d_in[7];
  const float* upd_b    = (const float*)d_in[8];
  const float* uln_g    = (const float*)d_in[9];
  const float* uln_b    = (const float*)d_in[10];
  float* out = (float*)d_out;

  float* ws = (float*)d_ws;
  size_t off = 0;
  float* S_g    = ws + off; off += (size_t)NN * NN;
  float* S_u    = ws + off; off += (size_t)NN * NN;
  float* ne_g   = ws + off; off += (size_t)NN * DD;
  float* ne_u   = ws + off; off += (size_t)NN * DD;
  float* bias_g = ws + off; off += (size_t)NN * OG;
  float* bias_u = ws + off; off += (size_t)NN * OU;
  float* Zb     = ws + off; off += (size_t)NN * NC;   // keep Zb,Y1,Y2 contiguous:
  float* Y1     = ws + off; off += (size_t)NN * NC;   // edge-block async overreads
  float* Y2     = ws + off; off += (size_t)NN * NC;   // stay inside the workspace
  float* zr     = ws + off; off += (size_t)NN * BB * OG;

  const int catBlocks = (NN * NC) / 256;                     // 16640
  const int mixBlocks = 8 * 33;                              // 264 macro-tiles
  const size_t ldsMix  = (size_t)(2 * A_DW + 2 * B_DW) * 4;  // 70656 B
  const size_t ldsGate = (size_t)(BB * KIP + KIP * OG) * 4;  // 150528 B
  const size_t ldsUpd  = (size_t)(BB * KIP + KIP * OU) * 4;  // 100352 B

  for (int t = 0; t < TT; ++t) {
    int pv = (t > 0);
    prep_kernel<<<NN, 128, 0, stream>>>(node_emb, time_emb + t * DD, gln_g, gln_b,
                                        uln_g, uln_b, gate_b, upd_b,
                                        ne_g, ne_u, bias_g, bias_u);
    adjacency_kernel<<<2 * NN, 256, 0, stream>>>(ne_g, ne_u, S_g, S_u);
    // ---- gate GCN ----
    build_cat_kernel<<<catBlocks, 256, 0, stream>>>(x, out, zr, Zb, t, 0, pv);
    gemm_mix_kernel<<<mixBlocks, 256, ldsMix, stream>>>(S_g, Zb, Zb, Y1, 0);
    gemm_mix_kernel<<<mixBlocks, 256, ldsMix, stream>>>(S_g, Y1, Zb, Y2, 1);
    pernode_gate_kernel<<<NN, 256, ldsGate, stream>>>(Zb, Y1, Y2, ne_g, gate_W,
                                                      bias_g, zr);
    // ---- update GCN + GRU blend ----
    build_cat_kernel<<<catBlocks, 256, 0, stream>>>(x, out, zr, Zb, t, 1, pv);
    gemm_mix_kernel<<<mixBlocks, 256, ldsMix, stream>>>(S_u, Zb, Zb, Y1, 0);
    gemm_mix_kernel<<<mixBlocks, 256, ldsMix, stream>>>(S_u, Y1, Zb, Y2, 1);
    pernode_update_kernel<<<NN, 256, ldsUpd, stream>>>(Zb, Y1, Y2, ne_u, upd_W,
                                                       bias_u, zr, out, t, pv);
  }
}